// MaskDICELoss_26980984553704
// MI455X (gfx1250) — compile-verified
//
#include <hip/hip_runtime.h>
#include <cstdint>

// ---------------- problem constants ----------------
#define BDIM 4
#define PDIM 65536
#define QDIM 300
#define QP1  301
#define ODIM 20
#define OP1  21
#define MDIM 20

// ---------------- tiling ----------------
#define ROWS 8                                   // pixel rows per chunk (1 per wave)
#define CHUNKS 16                                // chunks per block
#define BLOCKS_PER_B ((PDIM / ROWS) / CHUNKS)    // 512
#define NBLOCKS (BDIM * BLOCKS_PER_B)            // 2048
#define NTHREADS 256                             // 8 waves (wave32)
#define NWAVES (NTHREADS / 32)
#define SELW ((QP1 + 31) / 32)                   // 10 selmask words

#if defined(__has_builtin)
#if __has_builtin(__builtin_amdgcn_tensor_load_to_lds) && __has_builtin(__builtin_amdgcn_s_wait_tensorcnt)
#define HAVE_TDM 1
#endif
#endif
#ifndef HAVE_TDM
#define HAVE_TDM 0
#endif

typedef unsigned int v4u __attribute__((ext_vector_type(4)));
typedef int          v4i __attribute__((ext_vector_type(4)));
typedef int          v8i __attribute__((ext_vector_type(8)));

#if HAVE_TDM
// Issue one 2D TDM tile load: `rows` rows of `dim0` 4-byte elements,
// row stride == dim0 elements (tile contiguous in both memory and LDS).
// D# packing per CDNA5 ISA.
__device__ __forceinline__ void tdm_load_2d(uint32_t lds_addr, uint64_t gaddr,
                                            uint32_t dim0, uint32_t rows) {
  v4u g0;
  g0.x = 1u;                                                       // count=1 (valid D#)
  g0.y = lds_addr;                                                 // LDS byte address
  g0.z = (uint32_t)gaddr;                                          // global addr [31:0]
  g0.w = ((uint32_t)(gaddr >> 32) & 0x01FFFFFFu) | (2u << 30);     // addr[56:32] | type=2
  v8i g1;
  g1[0] = (int)(2u << 16);                                         // data_size=2 -> 4B; wg_mask=0
  g1[1] = (int)((dim0 & 0xFFFFu) << 16);                           // tensor_dim0[15:0]
  g1[2] = (int)((dim0 >> 16) | ((rows & 0xFFFFu) << 16));          // tensor_dim0[31:16]|tensor_dim1[15:0]
  g1[3] = (int)((rows >> 16) | ((dim0 & 0xFFFFu) << 16));          // tensor_dim1[31:16]|tile_dim0
  g1[4] = (int)(rows & 0xFFFFu);                                   // tile_dim1 | tile_dim2=0
  g1[5] = (int)dim0;                                               // tensor_dim0_stride[31:0]
  g1[6] = 0;                                                       // stride0[47:32]|stride1[15:0]
  g1[7] = 0;                                                       // stride1[47:16]
  v4i z4 = {0, 0, 0, 0};
#if defined(__clang_major__) && (__clang_major__ >= 23)
  v8i z8 = {0, 0, 0, 0, 0, 0, 0, 0};
  __builtin_amdgcn_tensor_load_to_lds(g0, g1, z4, z4, z8, 0);
#else
  __builtin_amdgcn_tensor_load_to_lds(g0, g1, z4, z4, 0);
#endif
}
#endif

__global__ __launch_bounds__(NTHREADS) void dice_zero(float* accum) {
  if (threadIdx.x < 2 * BDIM) accum[threadIdx.x] = 0.f;
}

__global__ __launch_bounds__(NTHREADS) void dice_main(
    const float* __restrict__ pred_logits,   // (B,P,Q+1) f32
    const float* __restrict__ q_logits,      // (B,Q)     f32
    const int*   __restrict__ pred_presence, // (B,P,Q+1) i32
    const int*   __restrict__ true_presence, // (B,P,O+1) i32
    const int*   __restrict__ matched,       // (B,2,M)   i32
    float*       __restrict__ accum)         // (2*B)     f32 partial num/den
{
  __shared__ float sL[2][ROWS * QP1];        // logits tiles (double buffered)
  __shared__ int   sP[2][ROWS * QP1];        // presence tiles
  __shared__ int   sT[2][ROWS * OP1];        // true-presence tiles
  __shared__ unsigned selmask[SELW];
  __shared__ unsigned char tbl[QP1];         // q -> inds1[m] (0xFF = unmatched)
  __shared__ float wacc[NWAVES][2];

  const int tid  = (int)threadIdx.x;
  const int wave = tid >> 5;
  const int lane = tid & 31;
  const int b      = (int)blockIdx.x / BLOCKS_PER_B;
  const int chunk0 = ((int)blockIdx.x % BLOCKS_PER_B) * CHUNKS;

#if HAVE_TDM
  auto issue = [&](int ci, int buf) {
    size_t rowbase = (size_t)b * PDIM + (size_t)(chunk0 + ci) * ROWS;
    tdm_load_2d((uint32_t)(uintptr_t)&sL[buf][0],
                (uint64_t)(uintptr_t)pred_logits + rowbase * (size_t)QP1 * 4u, QP1, ROWS);
    tdm_load_2d((uint32_t)(uintptr_t)&sP[buf][0],
                (uint64_t)(uintptr_t)pred_presence + rowbase * (size_t)QP1 * 4u, QP1, ROWS);
    tdm_load_2d((uint32_t)(uintptr_t)&sT[buf][0],
                (uint64_t)(uintptr_t)true_presence + rowbase * (size_t)OP1 * 4u, OP1, ROWS);
  };
  if (wave == 0) issue(0, 0);   // overlap first DMA with table setup
#endif

  // ---- per-b tables: sel bitmask and match table (301 entries, 256 threads!) ----
  if (tid < SELW) {
    unsigned w = 0;
    for (int j = 0; j < 32; ++j) {
      int q = tid * 32 + j;
      if (q == QDIM) w |= 1u << j;                                  // sel padded True at q=Q
      else if (q < QDIM && q_logits[(size_t)b * QDIM + q] > 0.f) w |= 1u << j;
    }
    selmask[tid] = w;
  }
  for (int q = tid; q < QP1; q += NTHREADS) tbl[q] = 0xFF;
  __syncthreads();
  if (tid < MDIM) {
    int q = matched[b * 2 * MDIM + tid];          // inds0[m] (distinct)
    int o = matched[b * 2 * MDIM + MDIM + tid];   // inds1[m] (permutation of O)
    tbl[q] = (unsigned char)o;
    atomicOr(&selmask[q >> 5], 1u << (q & 31));   // sel forced True at inds0
  }
  __syncthreads();

  float accN = 0.f, accD = 0.f;

  auto process = [&](int buf) {
    const float* Lr = &sL[buf][wave * QP1];
    const int*   Pr = &sP[buf][wave * QP1];
    // true-presence bits for this pixel (21 bits)
    int v = 1;
    if (lane < OP1) v = sT[buf][wave * OP1 + lane];
    unsigned tv = (unsigned)__ballot(v == 0);

    // pass 1: masked logits into registers + masked max
    float lreg[10];
    unsigned mbits = 0;
    float mx = -3.402823466e38f;
#pragma unroll
    for (int it = 0; it < 10; ++it) {
      int q = lane + 32 * it;
      float lv = 0.f; int pv = 1;
      if (q < QP1) { lv = Lr[q]; pv = Pr[q]; }
      bool mk = (q < QP1) && (pv == 0) && ((selmask[it] >> lane) & 1u);
      lreg[it] = lv;
      if (mk) { mbits |= (1u << it); mx = fmaxf(mx, lv); }
    }
#pragma unroll
    for (int off = 16; off; off >>= 1) mx = fmaxf(mx, __shfl_xor(mx, off, 32));

    // pass 2: exp + accumulate denom / matched-e / background-e
    float dn = 0.f, me = 0.f, bg = 0.f;
#pragma unroll
    for (int it = 0; it < 10; ++it) {
      if (mbits & (1u << it)) {
        int q = lane + 32 * it;
        float e = __expf(lreg[it] - mx);
        dn += e;
        unsigned char t = tbl[q];
        if (t < ODIM && ((tv >> t) & 1u)) me += e;
        if (q == QDIM) bg = e;
      }
    }
#pragma unroll
    for (int off = 16; off; off >>= 1) {
      dn += __shfl_xor(dn, off, 32);
      me += __shfl_xor(me, off, 32);
      bg += __shfl_xor(bg, off, 32);
    }

    if (lane == 0) {
      float popc = (float)__popc((int)(tv & 0xFFFFFu));   // sum_m true_m (inds1 is a permutation)
      float tv20 = (float)((tv >> 20) & 1u);              // true background bit
      if (dn > 0.f) {                                     // cover
        accN += 2.f * (me + bg * tv20) / dn;              // 2*(sum pred_m*true_m + pred_bg*true_bg)
        accD += 1.f + popc + tv20;                        // sum(soft)=1 + sum(true_m) + true_bg
      } else {
        accD += popc;                                     // true_m counted even when uncovered
      }
    }
  };

#if HAVE_TDM
  for (int i = 0; i < CHUNKS; ++i) {
    if (wave == 0) {
      if (i + 1 < CHUNKS) {
        issue(i + 1, (i + 1) & 1);                 // prefetch next chunk (other buffer)
        __builtin_amdgcn_s_wait_tensorcnt(3);      // chunk i's 3 DMAs complete (in-order)
      } else {
        __builtin_amdgcn_s_wait_tensorcnt(0);
      }
    }
    __syncthreads();
    process(i & 1);
    __syncthreads();
  }
#else
  for (int i = 0; i < CHUNKS; ++i) {
    size_t rowbase = (size_t)b * PDIM + (size_t)(chunk0 + i) * ROWS;
    const float* gl = pred_logits + rowbase * QP1;
    const int*   gp = pred_presence + rowbase * QP1;
    const int*   gt = true_presence + rowbase * OP1;
    for (int j = tid; j < ROWS * QP1; j += NTHREADS) { sL[0][j] = gl[j]; sP[0][j] = gp[j]; }
    for (int j = tid; j < ROWS * OP1; j += NTHREADS) sT[0][j] = gt[j];
    __syncthreads();
    process(0);
    __syncthreads();
  }
#endif

  if (lane == 0) { wacc[wave][0] = accN; wacc[wave][1] = accD; }
  __syncthreads();
  if (tid == 0) {
    float n = 0.f, d = 0.f;
    for (int w = 0; w < NWAVES; ++w) { n += wacc[w][0]; d += wacc[w][1]; }
    atomicAdd(&accum[b], n);
    atomicAdd(&accum[BDIM + b], d);
  }
}

__global__ void dice_finalize(const float* __restrict__ accum, float* __restrict__ out) {
  if (threadIdx.x == 0 && blockIdx.x == 0) {
    float s = 0.f;
    for (int bb = 0; bb < BDIM; ++bb)
      s += 1.f - (accum[bb] + 1.f) / (accum[BDIM + bb] + 1.f);
    out[0] = s * (1.f / BDIM);
  }
}

extern "C" void kernel_launch(void* const* d_in, const int* in_sizes, int n_in,
                              void* d_out, int out_size, void* d_ws, size_t ws_size,
                              hipStream_t stream) {
  (void)in_sizes; (void)n_in; (void)out_size; (void)ws_size;
  const float* pred_logits   = (const float*)d_in[0];
  const float* q_logits      = (const float*)d_in[1];
  const int*   pred_presence = (const int*)d_in[2];
  const int*   true_presence = (const int*)d_in[3];
  const int*   matched       = (const int*)d_in[4];
  float* out   = (float*)d_out;
  float* accum = (float*)d_ws;   // 2*B floats of scratch

  dice_zero<<<1, NTHREADS, 0, stream>>>(accum);
  dice_main<<<dim3(NBLOCKS), dim3(NTHREADS), 0, stream>>>(
      pred_logits, q_logits, pred_presence, true_presence, matched, accum);
  dice_finalize<<<1, 1, 0, stream>>>(accum, out);
}